// GCNEncoder_9268539425058
// MI455X (gfx1250) — compile-verified
//
#include <hip/hip_runtime.h>
#include <math.h>

typedef __attribute__((ext_vector_type(2))) float v2f;
typedef __attribute__((ext_vector_type(8))) float v8f;

// ---------------------------------------------------------------------------
// Degree / normalization kernels
// ---------------------------------------------------------------------------
__global__ void k_init_deg(float* deg, int N) {
    int i = blockIdx.x * blockDim.x + threadIdx.x;
    if (i < N) deg[i] = 1.0f;  // self-loop contributes 1 to every node's degree
}

__global__ void k_count_deg(const long long* __restrict__ col, float* deg, int E) {
    int e = blockIdx.x * blockDim.x + threadIdx.x;
    if (e < E) atomicAdd(&deg[(int)col[e]], 1.0f);
}

__global__ void k_finalize_dinv(float* deg, int N) {
    int i = blockIdx.x * blockDim.x + threadIdx.x;
    if (i < N) {
        float d = deg[i];
        deg[i] = (d > 0.0f) ? rsqrtf(d) : 0.0f;  // in place: deg -> dinv
    }
}

// ---------------------------------------------------------------------------
// WT[n*K + k] = W[k*Nc + n]  (one-shot, tiny: 16K / 8K elements)
// ---------------------------------------------------------------------------
__global__ void k_transpose(const float* __restrict__ W, float* __restrict__ WT,
                            int K, int Nc) {
    int t = blockIdx.x * blockDim.x + threadIdx.x;
    if (t >= K * Nc) return;
    int k = t / Nc, n = t - k * Nc;
    WT[(size_t)n * K + k] = W[t];
}

// ---------------------------------------------------------------------------
// f32 WMMA GEMM: Out[M x Nc] = act(A[M x K]) @ W[K x Nc], W given transposed
// as WT[Nc x K]. One wave computes one 16x16 output tile using
// V_WMMA_F32_16X16X4_F32 (K=4 per instruction, full f32 precision).
// A layout (ISA 7.12.2, 32-bit A 16x4): lane = (k>=2?16:0)+m, vgpr = k&1.
// With WT, the B operand uses the identical per-lane pattern (lane = n).
// ---------------------------------------------------------------------------
template <bool RELU_A>
__global__ void k_gemm_wmma_f32(const float* __restrict__ A,
                                const float* __restrict__ WT,
                                float* __restrict__ Out,
                                int M, int K, int Nc) {
    const int tilesN = Nc >> 4;
    const int waveId = (blockIdx.x * blockDim.x + threadIdx.x) >> 5;  // wave-uniform
    const int tilesM = (M + 15) >> 4;
    if (waveId >= tilesM * tilesN) return;  // uniform per wave: EXEC stays all-1s

    const int tileM = waveId / tilesN;
    const int tileN = waveId % tilesN;
    const int lane  = threadIdx.x & 31;
    const int m     = lane & 15;          // row within tile (A) / col within tile (B)
    const int half  = lane >> 4;          // 0: K-pair {0,1}; 1: K-pair {2,3}
    const int kOff  = half << 1;

    // clamp A row so partial M tiles load in-bounds with full EXEC
    int aRow = tileM * 16 + m;
    if (aRow > M - 1) aRow = M - 1;
    const float* aPtr = A + (size_t)aRow * K + kOff;
    const float* bPtr = WT + (size_t)(tileN * 16 + m) * K + kOff;

    v8f c = {};
    for (int k = 0; k < K; k += 4) {
        // A: two consecutive f32 (k+kOff, k+kOff+1) for this lane's row
        v2f a = *(const v2f*)(aPtr + k);
        if (RELU_A) {
            a.x = fmaxf(a.x, 0.0f);
            a.y = fmaxf(a.y, 0.0f);
        }
        // B: WT[n][k+kOff .. k+kOff+1] — contiguous b64 load, n = lane&15
        v2f b = *(const v2f*)(bPtr + k);
        // D = A*B + C  (v_wmma_f32_16x16x4_f32)
        c = __builtin_amdgcn_wmma_f32_16x16x4_f32(
            /*neg_a=*/false, a, /*neg_b=*/false, b,
            /*c_mod=*/(short)0, c, /*reuse_a=*/false, /*reuse_b=*/false);
    }

    // store: VGPR i -> row m = i + half*8, col n = lane&15
    const int n = tileN * 16 + m;
#pragma unroll
    for (int i = 0; i < 8; ++i) {
        int gRow = tileM * 16 + i + (half << 3);
        if (gRow < M) Out[(size_t)gRow * Nc + n] = c[i];
    }
}

// ---------------------------------------------------------------------------
// out[v][f] = h[v][f] * dinv[v]^2 + bias[f]   (self-loop term + bias)
// ---------------------------------------------------------------------------
__global__ void k_init_out(const float* __restrict__ h,
                           const float* __restrict__ dinv,
                           const float* __restrict__ bias,
                           float* __restrict__ out, int N, int F) {
    int t = blockIdx.x * blockDim.x + threadIdx.x;
    if (t >= N * F) return;
    int v = t / F, f = t - v * F;
    float di = dinv[v];
    out[t] = h[t] * di * di + bias[f];
}

// ---------------------------------------------------------------------------
// Edge scatter: out[col] += h[row] * dinv[row]*dinv[col]
// One thread per (edge, 4-feature chunk): float4 gather + 4 f32 atomics.
// A wave covers one edge's full feature row (F=128) -> index/dinv loads are
// same-address within the wave (single coalesced fetch).
// ---------------------------------------------------------------------------
__global__ void k_scatter(const float* __restrict__ h,
                          const long long* __restrict__ row,
                          const long long* __restrict__ col,
                          const float* __restrict__ dinv,
                          float* out, int E, int F) {
    const int Fd4 = F >> 2;
    int t = blockIdx.x * blockDim.x + threadIdx.x;
    if (t >= E * Fd4) return;
    int e  = t / Fd4;
    int ch = t - e * Fd4;
    int r  = (int)row[e];
    int cl = (int)col[e];
    float w = dinv[r] * dinv[cl];
    const float4 hv = ((const float4*)(h + (size_t)r * F))[ch];
    float* o = out + (size_t)cl * F + (ch << 2);
    atomicAdd(o + 0, hv.x * w);
    atomicAdd(o + 1, hv.y * w);
    atomicAdd(o + 2, hv.z * w);
    atomicAdd(o + 3, hv.w * w);
}

// ---------------------------------------------------------------------------
// Host-side launcher
// ---------------------------------------------------------------------------
extern "C" void kernel_launch(void* const* d_in, const int* in_sizes, int n_in,
                              void* d_out, int out_size, void* d_ws, size_t ws_size,
                              hipStream_t stream) {
    const float*     x    = (const float*)d_in[0];
    const long long* ei   = (const long long*)d_in[1];  // [2, E] int64, row-major
    const float*     W1   = (const float*)d_in[2];
    const float*     b1   = (const float*)d_in[3];
    const float*     W2   = (const float*)d_in[4];
    const float*     b2   = (const float*)d_in[5];
    float*           outp = (float*)d_out;

    const int Fin    = 128;
    const int N      = in_sizes[0] / Fin;         // 50000
    const int E      = in_sizes[1] / 2;           // 800000
    const int hidden = in_sizes[3];               // 128
    const int Fout   = in_sizes[5];               // 64

    const long long* srcs = ei;       // edge_index[0] = source (row)
    const long long* tgts = ei + E;   // edge_index[1] = target (col)

    // workspace layout (floats), 256-element aligned chunks
    float* wsf  = (float*)d_ws;
    size_t off  = 0;
    float* dinv = wsf + off; off += (size_t)((N + 255) & ~255);
    float* W1T  = wsf + off; off += (size_t)Fin * hidden;
    float* W2T  = wsf + off; off += (size_t)hidden * Fout;
    float* h1   = wsf + off; off += (size_t)N * hidden;
    float* agg1 = wsf + off; off += (size_t)N * hidden;
    float* h2   = wsf + off; off += (size_t)N * Fout;
    (void)ws_size;

    const int B = 256;
    auto cdiv = [](long long a, long long b) { return (int)((a + b - 1) / b); };

    // 0) transpose weights once (tiny)
    k_transpose<<<cdiv((long long)Fin * hidden, B), B, 0, stream>>>(W1, W1T, Fin, hidden);
    k_transpose<<<cdiv((long long)hidden * Fout, B), B, 0, stream>>>(W2, W2T, hidden, Fout);

    // 1) degrees -> dinv
    k_init_deg<<<cdiv(N, B), B, 0, stream>>>(dinv, N);
    k_count_deg<<<cdiv(E, B), B, 0, stream>>>(tgts, dinv, E);
    k_finalize_dinv<<<cdiv(N, B), B, 0, stream>>>(dinv, N);

    // 2) GEMM1: h1 = x @ W1   [N x 128] = [N x 128][128 x 128]
    {
        int tilesM = (N + 15) / 16, tilesN = hidden / 16;
        long long waves = (long long)tilesM * tilesN;
        k_gemm_wmma_f32<false><<<cdiv(waves * 32, B), B, 0, stream>>>(
            x, W1T, h1, N, Fin, hidden);
    }

    // 3) layer-1 aggregation: agg1 = selfloop + bias, then edge scatter
    k_init_out<<<cdiv((long long)N * hidden, B), B, 0, stream>>>(h1, dinv, b1, agg1, N, hidden);
    k_scatter<<<cdiv((long long)E * (hidden / 4), B), B, 0, stream>>>(
        h1, srcs, tgts, dinv, agg1, E, hidden);

    // 4) GEMM2 (ReLU fused into A load): h2 = relu(agg1) @ W2
    {
        int tilesM = (N + 15) / 16, tilesN = Fout / 16;
        long long waves = (long long)tilesM * tilesN;
        k_gemm_wmma_f32<true><<<cdiv(waves * 32, B), B, 0, stream>>>(
            agg1, W2T, h2, N, hidden, Fout);
    }

    // 5) layer-2 aggregation into d_out
    k_init_out<<<cdiv((long long)N * Fout, B), B, 0, stream>>>(h2, dinv, b2, outp, N, Fout);
    k_scatter<<<cdiv((long long)E * (Fout / 4), B), B, 0, stream>>>(
        h2, srcs, tgts, dinv, outp, E, Fout);
}